// GINLayer_81844896792885
// MI455X (gfx1250) — compile-verified
//
#include <hip/hip_runtime.h>

// GIN layer for MI455X (gfx1250, wave32).
// D = 96 (6 x 16 WMMA column tiles), N = 50000 (3125 x 16 row tiles).

#define DIM 96
#define TILES_N (DIM / 16)

typedef __attribute__((ext_vector_type(2))) float v2f;
typedef __attribute__((ext_vector_type(8))) float v8f;

// ---------------------------------------------------------------------------
// Kernel 1: pooled = (1 + eps) * feature ; also zero the BN stats buffer.
// ---------------------------------------------------------------------------
__global__ void gin_init(const float* __restrict__ feat,
                         const float* __restrict__ eps,
                         float* __restrict__ pooled,
                         float* __restrict__ stats,
                         int total) {
    int i = blockIdx.x * blockDim.x + threadIdx.x;
    if (i < 2 * DIM) stats[i] = 0.0f;           // sum[96] + sumsq[96]
    if (i >= total) return;
    pooled[i] = (1.0f + eps[0]) * feat[i];
}

// ---------------------------------------------------------------------------
// Kernel 2: edge scatter-add. One wave32 per edge; 3 iterations cover D=96.
// Gather feature[src[e]] (hits L2: feature table is 19MB << 192MB L2),
// atomicAdd into pooled[dst[e]].
// ---------------------------------------------------------------------------
__global__ void gin_scatter(const float* __restrict__ feat,
                            const int* __restrict__ src,
                            const int* __restrict__ dst,
                            float* __restrict__ pooled,
                            int nE) {
    int e    = (blockIdx.x * blockDim.x + threadIdx.x) >> 5;
    int lane = threadIdx.x & 31;
    if (e >= nE) return;
    int s = src[e];
    int d = dst[e];
    const float* f = feat   + (size_t)s * DIM;
    float*       p = pooled + (size_t)d * DIM;
#pragma unroll
    for (int i = 0; i < 3; ++i) {
        atomicAdd(&p[lane + 32 * i], f[lane + 32 * i]);
    }
}

// ---------------------------------------------------------------------------
// Kernel 3/4: C = act(A @ W + bias), one wave per 16x16 output tile using
// V_WMMA_F32_16X16X4_F32 (fp32, K=4 per instruction, 24 steps over K=96).
//
// A (16x4) lane layout: lane l -> row (l&15), K pair ((l>>4)*2, +1)  [2 VGPRs]
// B (4x16) lane layout: lane l -> col n0+(l&15), same K pair          [2 VGPRs]
// C/D (16x16): VGPR r -> row row0 + (l>>4)*8 + r, col n0 + (l&15)    [8 VGPRs]
//
// Tile index check is wave-uniform, so EXEC is all-ones at the WMMA
// (ISA requirement, CDNA5 ISA).
// ---------------------------------------------------------------------------
template <bool RELU>
__global__ void gin_gemm(const float* __restrict__ A,
                         const float* __restrict__ W,
                         const float* __restrict__ bias,
                         float* __restrict__ out,
                         int nTiles, int nRows) {
    int tile = blockIdx.x * (blockDim.x >> 5) + (threadIdx.x >> 5);
    if (tile >= nTiles) return;                  // wave-uniform
    const int lane = threadIdx.x & 31;
    const int tm   = tile / TILES_N;
    const int tn   = tile % TILES_N;
    const int row0 = tm * 16;
    const int n0   = tn * 16;
    const int m    = lane & 15;
    const int kk   = (lane >> 4) << 1;          // 0 or 2

    const float* arow  = A + (size_t)(row0 + m) * DIM + kk;          // contiguous K pair
    const float* wbase = W + (size_t)kk * DIM + n0 + m;              // strided rows of W

    v8f c = {};
#pragma unroll
    for (int k0 = 0; k0 < DIM; k0 += 4) {
        v2f a = *(const v2f*)(arow + k0);        // 8B-aligned (kk even, k0 % 4 == 0)
        v2f b;
        b.x = wbase[(size_t)k0 * DIM];
        b.y = wbase[(size_t)(k0 + 1) * DIM];
        // (neg_a, A, neg_b, B, c_mod, C, reuse_a, reuse_b)
        c = __builtin_amdgcn_wmma_f32_16x16x4_f32(
                false, a, false, b, (short)0, c, false, false);
    }

    const int   nc = n0 + m;
    const float bv = bias[nc];
    const int   rbase = row0 + (lane >> 4) * 8;
#pragma unroll
    for (int r = 0; r < 8; ++r) {
        if (rbase + r < nRows) {
            float v = c[r] + bv;
            if (RELU) v = fmaxf(v, 0.0f);
            out[(size_t)(rbase + r) * DIM + nc] = v;
        }
    }
}

// ---------------------------------------------------------------------------
// Kernel 5: per-column sum / sum-of-squares. Grid stride is a multiple of 96
// so each thread accumulates one column privately, then one atomicAdd pair.
// ---------------------------------------------------------------------------
__global__ void gin_colreduce(const float* __restrict__ h,
                              float* __restrict__ stats,
                              int total) {
    int gid    = blockIdx.x * blockDim.x + threadIdx.x;
    int stride = gridDim.x * blockDim.x;   // launch chosen so stride % 96 == 0
    int col    = gid % DIM;
    float s = 0.0f, sq = 0.0f;
    for (int i = gid; i < total; i += stride) {
        float v = h[i];
        s  += v;
        sq += v * v;
    }
    atomicAdd(&stats[col], s);
    atomicAdd(&stats[DIM + col], sq);
}

// ---------------------------------------------------------------------------
// Kernel 6: fold mean/var/gamma/beta into per-column scale & shift.
// ---------------------------------------------------------------------------
__global__ void gin_bnfinalize(const float* __restrict__ stats,
                               const float* __restrict__ gamma,
                               const float* __restrict__ beta,
                               float* __restrict__ coef,
                               float invN) {
    int c = threadIdx.x;
    if (c >= DIM) return;
    float mean = stats[c] * invN;
    float var  = stats[DIM + c] * invN - mean * mean;
    float sc   = gamma[c] * rsqrtf(var + 1e-5f);
    coef[c]        = sc;
    coef[DIM + c]  = beta[c] - mean * sc;
}

// ---------------------------------------------------------------------------
// Kernel 7: in-place BN apply + ReLU on d_out.
// ---------------------------------------------------------------------------
__global__ void gin_bnapply(float* __restrict__ h,
                            const float* __restrict__ coef,
                            int total) {
    int i = blockIdx.x * blockDim.x + threadIdx.x;
    if (i >= total) return;
    int c = i % DIM;
    float v = h[i] * coef[c] + coef[DIM + c];
    h[i] = fmaxf(v, 0.0f);
}

// ---------------------------------------------------------------------------
extern "C" void kernel_launch(void* const* d_in, const int* in_sizes, int n_in,
                              void* d_out, int out_size, void* d_ws, size_t ws_size,
                              hipStream_t stream) {
    const float* feat  = (const float*)d_in[0];
    const int*   src   = (const int*)  d_in[1];
    const int*   dst   = (const int*)  d_in[2];
    const float* W1    = (const float*)d_in[3];
    const float* b1    = (const float*)d_in[4];
    const float* W2    = (const float*)d_in[5];
    const float* b2    = (const float*)d_in[6];
    const float* gamma = (const float*)d_in[7];
    const float* beta  = (const float*)d_in[8];
    const float* eps   = (const float*)d_in[9];
    float*       out   = (float*)d_out;

    const int N     = in_sizes[0] / DIM;   // 50000
    const int E     = in_sizes[1];         // 800000
    const int total = N * DIM;

    // Workspace layout: pooled[N*D] | h1[N*D] | stats[2*D] | coef[2*D]
    float* pooled = (float*)d_ws;
    float* h1     = pooled + (size_t)total;
    float* stats  = h1 + (size_t)total;
    float* coef   = stats + 2 * DIM;

    // 1. pooled = (1+eps)*feature, zero stats
    gin_init<<<(total + 255) / 256, 256, 0, stream>>>(feat, eps, pooled, stats, total);

    // 2. scatter-add over edges (1 wave per edge, 8 waves / block)
    gin_scatter<<<(E + 7) / 8, 256, 0, stream>>>(feat, src, dst, pooled, E);

    // 3/4. WMMA GEMMs
    const int nTiles = ((N + 15) / 16) * TILES_N;
    const int gblk   = (nTiles + 7) / 8;           // 8 waves per 256-thread block
    gin_gemm<true ><<<gblk, 256, 0, stream>>>(pooled, W1, b1, h1,  nTiles, N);
    gin_gemm<false><<<gblk, 256, 0, stream>>>(h1,     W2, b2, out, nTiles, N);

    // 5. column reduce: 300 blocks * 256 = 76800 threads (multiple of 96)
    gin_colreduce<<<300, 256, 0, stream>>>(out, stats, total);

    // 6. BN coefficients
    gin_bnfinalize<<<1, 128, 0, stream>>>(stats, gamma, beta, coef, 1.0f / (float)N);

    // 7. normalize + ReLU in place
    gin_bnapply<<<(total + 255) / 256, 256, 0, stream>>>(out, coef, total);
}